// _CompiledModel_71055938945281
// MI455X (gfx1250) — compile-verified
//
#include <hip/hip_runtime.h>
#include <math.h>

// MI455X / gfx1250 fused periodic pair-energy + force kernel.
// Distance tiles via V_WMMA_F32_16X16X4_F32:
//   A[M][:] = [-2x_i, -2y_i, -2z_i, 1]        (16x4, rows = local atoms)
//   B[:][N] = [ x_j,   y_j,   z_j, |x_j|^2 ]  (4x16, cols = ext atoms)
//   C[M][N] = |x_i|^2
//   => D[M][N] = d2(i,j)   (one WMMA per 256 candidate pairs)
// Hot loop is register-resident: B built straight from a per-lane global
// load (lane's output column is fixed), row data preloaded to registers,
// LDS only holds the dynamically-indexed 4x4 pair table.

#define NLOC   2048
#define NCELL  27
#define NALL   (NLOC * NCELL)
#define RCUT_F 6.0f
#define RCUT2  (RCUT_F * RCUT_F)
#define NTYPES 4

typedef __attribute__((ext_vector_type(2))) float v2f;
typedef __attribute__((ext_vector_type(8))) float v8f;

// ---------------------------------------------------------------------------
// Kernel A: wrap coords into the box, emit 27 shifted images {x,y,z,|x|^2}.
// Image c=0 is the (0,0,0) shift (needed for self-pair exclusion).
// ---------------------------------------------------------------------------
__global__ void build_ext_kernel(const float* __restrict__ coord,
                                 const float* __restrict__ box,
                                 float4* __restrict__ ext4)
{
  int i = blockIdx.x * blockDim.x + threadIdx.x;
  if (i >= NLOC) return;

  float b00 = box[0], b01 = box[1], b02 = box[2];
  float b10 = box[3], b11 = box[4], b12 = box[5];
  float b20 = box[6], b21 = box[7], b22 = box[8];

  float c00 =  (b11 * b22 - b12 * b21);
  float c01 = -(b10 * b22 - b12 * b20);
  float c02 =  (b10 * b21 - b11 * b20);
  float det = b00 * c00 + b01 * c01 + b02 * c02;
  float id  = 1.0f / det;
  float i00 =  (b11 * b22 - b12 * b21) * id;
  float i01 = -(b01 * b22 - b02 * b21) * id;
  float i02 =  (b01 * b12 - b02 * b11) * id;
  float i10 = -(b10 * b22 - b12 * b20) * id;
  float i11 =  (b00 * b22 - b02 * b20) * id;
  float i12 = -(b00 * b12 - b02 * b10) * id;
  float i20 =  (b10 * b21 - b11 * b20) * id;
  float i21 = -(b00 * b21 - b01 * b20) * id;
  float i22 =  (b00 * b11 - b01 * b10) * id;

  float x = coord[3 * i + 0], y = coord[3 * i + 1], z = coord[3 * i + 2];
  float f0 = x * i00 + y * i10 + z * i20;
  float f1 = x * i01 + y * i11 + z * i21;
  float f2 = x * i02 + y * i12 + z * i22;
  f0 -= floorf(f0); f1 -= floorf(f1); f2 -= floorf(f2);
  float cx = f0 * b00 + f1 * b10 + f2 * b20;
  float cy = f0 * b01 + f1 * b11 + f2 * b21;
  float cz = f0 * b02 + f1 * b12 + f2 * b22;

  #pragma unroll 1
  for (int c = 0; c < NCELL; ++c) {
    int idx = (c == 0) ? 13 : (c <= 13 ? c - 1 : c);
    float sx = (float)(idx / 9 - 1);
    float sy = (float)((idx / 3) % 3 - 1);
    float sz = (float)(idx % 3 - 1);
    float px = cx + sx * b00 + sy * b10 + sz * b20;
    float py = cy + sx * b01 + sy * b11 + sz * b21;
    float pz = cz + sx * b02 + sy * b12 + sz * b22;
    ext4[c * NLOC + i] = make_float4(px, py, pz, px * px + py * py + pz * pz);
  }
}

// ---------------------------------------------------------------------------
// Kernel B: one wave32 per (16-atom row tile, image chunk).
// ---------------------------------------------------------------------------
__global__ __launch_bounds__(32)
void pair_tile_kernel(const float4* __restrict__ ext4,
                      const int*    __restrict__ atype,
                      const float*  __restrict__ table,
                      float4*       __restrict__ partial)
{
  const int lane  = threadIdx.x;      // 0..31 (wave32)
  const int rowT  = blockIdx.x;       // 0..127
  const int chunk = blockIdx.y;       // 0..26
  const int row0  = rowT * 16;

  __shared__ float4 sXi[16];
  __shared__ int    sTi[16];
  __shared__ float  sTab[NTYPES * NTYPES];

  if (lane < 16) {
    sXi[lane]  = ext4[row0 + lane];   // chunk 0 = wrapped local coords
    sTi[lane]  = atype[row0 + lane];
    sTab[lane] = table[lane];
  }
  __syncthreads();

  const int half  = lane >> 4;        // 0: K=0,1 ; 1: K=2,3
  const int mrow  = lane & 15;        // lane's A-row / output column N
  const int mbase = half * 8;         // C/D rows: vgpr r -> M = r + mbase

  // A matrix (loop-invariant), rows striped across lanes.
  float4 xr = sXi[mrow];
  v2f A;
  A[0] = half ? (-2.0f * xr.z) : (-2.0f * xr.x);
  A[1] = half ? 1.0f           : (-2.0f * xr.y);

  // Register-resident row data for this lane's 8 output rows.
  v8f C;
  float4 xiR[8];
  int    tiR[8];
  #pragma unroll
  for (int r = 0; r < 8; ++r) {
    xiR[r] = sXi[mbase + r];
    tiR[r] = sTi[mbase + r];
    C[r]   = xiR[r].w;                // |x_i|^2
  }

  float accE[8], accFx[8], accFy[8], accFz[8];
  #pragma unroll
  for (int r = 0; r < 8; ++r) { accE[r] = 0.f; accFx[r] = 0.f; accFy[r] = 0.f; accFz[r] = 0.f; }

  const float pi_rc   = 3.14159265358979323846f / RCUT_F;
  const int   jchunk0 = chunk * NLOC;

  #pragma unroll 1
  for (int t = 0; t < NLOC / 16; ++t) {
    const int jl = t * 16 + mrow;               // ext atom local id for this lane's column
    float4 xj = ext4[jchunk0 + jl];             // coalesced, L2-resident
    int    tj = atype[jl];

    v2f B;                                      // columns striped across lanes
    B[0] = half ? xj.z : xj.x;
    B[1] = half ? xj.w : xj.y;                  // K=3 carries |x_j|^2

    v8f D = __builtin_amdgcn_wmma_f32_16x16x4_f32(
        false, A, false, B, (short)0, C, false, false);

    // Wave-uniform early-out: ~88% of tiles have no pair inside RCUT.
    bool hit = false;
    #pragma unroll
    for (int r = 0; r < 8; ++r) hit = hit || (D[r] < RCUT2);
    if (!__any(hit)) continue;

    #pragma unroll
    for (int r = 0; r < 8; ++r) {
      float d2 = fmaxf(D[r], 0.0f);
      int   M    = mbase + r;
      bool  self = (chunk == 0) && (jl == row0 + M);
      if (d2 < RCUT2 && !self) {
        float d2e  = d2 + 1e-12f;
        float rinv = __builtin_amdgcn_rsqf(d2e);   // v_rsq_f32: no div/sqrt chains
        float rr   = d2e * rinv;
        float arg  = pi_rc * rr;
        float sw   = 0.5f * (__cosf(arg) + 1.0f);
        float swp  = -0.5f * pi_rc * __sinf(arg);
        float u    = sw * rinv;                    // sw/r
        float up   = (swp - u) * rinv;             // d/dr (sw/r)
        int   ti  = tiR[r];
        float aKJ = sTab[ti * NTYPES + tj];
        float aJK = sTab[tj * NTYPES + ti];
        accE[r] += aKJ * u;
        // Newton-symmetric force: F_K += 0.5(a_KJ + a_JK) u'(r) (x_j - x_K)/r
        float coef = 0.5f * (aKJ + aJK) * up * rinv;
        accFx[r] += coef * (xj.x - xiR[r].x);
        accFy[r] += coef * (xj.y - xiR[r].y);
        accFz[r] += coef * (xj.z - xiR[r].z);
      }
    }
  }

  // Deterministic fixed-order reduction over the 16 lanes of each half
  // (xor offsets 8,4,2,1 never cross the lane&16 boundary).
  #pragma unroll
  for (int r = 0; r < 8; ++r) {
    float e = accE[r], fx = accFx[r], fy = accFy[r], fz = accFz[r];
    #pragma unroll
    for (int off = 8; off >= 1; off >>= 1) {
      e  += __shfl_xor(e,  off, 32);
      fx += __shfl_xor(fx, off, 32);
      fy += __shfl_xor(fy, off, 32);
      fz += __shfl_xor(fz, off, 32);
    }
    if (mrow == 0)
      partial[chunk * NLOC + row0 + mbase + r] = make_float4(e, fx, fy, fz);
  }
}

// ---------------------------------------------------------------------------
// Kernel C: fold 27 per-image partials per atom, fixed order.
// Output layout: [energy(1) | atom_energy(2048) | force(2048*3)]
// ---------------------------------------------------------------------------
__global__ void finalize_kernel(const float4* __restrict__ partial,
                                float* __restrict__ out)
{
  int i = blockIdx.x * blockDim.x + threadIdx.x;
  if (i >= NLOC) return;
  float e = 0.f, fx = 0.f, fy = 0.f, fz = 0.f;
  #pragma unroll 1
  for (int c = 0; c < NCELL; ++c) {
    float4 p = partial[c * NLOC + i];
    e += p.x; fx += p.y; fy += p.z; fz += p.w;
  }
  out[1 + i] = 0.5f * e;
  out[1 + NLOC + 3 * i + 0] = fx;
  out[1 + NLOC + 3 * i + 1] = fy;
  out[1 + NLOC + 3 * i + 2] = fz;
}

// ---------------------------------------------------------------------------
// Kernel D: deterministic single-block tree reduction of atom energies.
// ---------------------------------------------------------------------------
__global__ __launch_bounds__(256)
void energy_reduce_kernel(float* __restrict__ out)
{
  __shared__ float s[256];
  int t = threadIdx.x;
  float acc = 0.f;
  for (int i = t; i < NLOC; i += 256) acc += out[1 + i];
  s[t] = acc;
  __syncthreads();
  for (int w = 128; w > 0; w >>= 1) {
    if (t < w) s[t] += s[t + w];
    __syncthreads();
  }
  if (t == 0) out[0] = s[0];
}

// ---------------------------------------------------------------------------
extern "C" void kernel_launch(void* const* d_in, const int* in_sizes, int n_in,
                              void* d_out, int out_size, void* d_ws, size_t ws_size,
                              hipStream_t stream)
{
  const float* coord = (const float*)d_in[0];   // (1, 2048, 3) f32
  const int*   atype = (const int*)  d_in[1];   // (1, 2048)    i32
  const float* box   = (const float*)d_in[2];   // (1, 3, 3)    f32
  const float* table = (const float*)d_in[3];   // (4, 4)       f32
  float* out = (float*)d_out;

  char* ws = (char*)d_ws;
  float4* ext4    = (float4*)ws;                                   // 55296 * 16B
  float4* partial = (float4*)(ws + (size_t)NALL * sizeof(float4)); // 55296 * 16B

  build_ext_kernel<<<(NLOC + 255) / 256, 256, 0, stream>>>(coord, box, ext4);
  pair_tile_kernel<<<dim3(NLOC / 16, NCELL), 32, 0, stream>>>(ext4, atype, table, partial);
  finalize_kernel<<<(NLOC + 255) / 256, 256, 0, stream>>>(partial, out);
  energy_reduce_kernel<<<1, 256, 0, stream>>>(out);

  (void)in_sizes; (void)n_in; (void)out_size; (void)ws_size;
}